// FWHT_13322988552690
// MI455X (gfx1250) — compile-verified
//
#include <hip/hip_runtime.h>

typedef __attribute__((ext_vector_type(2))) float v2f;
typedef __attribute__((ext_vector_type(8))) float v8f;

#define FWHT_D      4096
#define FWHT_WAVES  4
#define FWHT_TPB    (FWHT_WAVES * 32)
// pad one float per 16 to kill stride-16 / stride-256 bank conflicts (17 coprime to 64 banks)
#define PADIDX(i)   ((i) + ((i) >> 4))
#define ROW_PADDED  (FWHT_D + (FWHT_D >> 4))   // 4352 floats = 17408 B per row

__global__ __launch_bounds__(FWHT_TPB) void fwht4096_wmma_kernel(
    const float* __restrict__ x, float* __restrict__ y, int nrows) {
  __shared__ __align__(16) float lds[FWHT_WAVES * ROW_PADDED];

  const int lane = threadIdx.x & 31;
  const int wave = threadIdx.x >> 5;
  const int row  = blockIdx.x * FWHT_WAVES + wave;
  const bool valid = (row < nrows);

  float* __restrict__ L = lds + wave * ROW_PADDED;
  const float* __restrict__ xr = x + (size_t)row * FWHT_D;
  float*       __restrict__ yr = y + (size_t)row * FWHT_D;

  // ---- stage row into LDS: async DMA, no VGPR round-trip (ASYNCcnt-tracked) ----
  if (valid) {
#pragma unroll
    for (int i = 0; i < FWHT_D / (32 * 4); ++i) {
      const int e = (i * 32 + lane) * 4;              // multiple of 4 -> same pad for all 4
      const uint32_t dst = (uint32_t)(uintptr_t)(L + PADIDX(e));  // LDS byte address
      const uint64_t src = (uint64_t)(uintptr_t)(xr + e);
      asm volatile("global_load_async_to_lds_b128 %0, %1, off"
                   :: "v"(dst), "v"(src) : "memory");
    }
  }
  asm volatile("s_wait_asynccnt 0x0" ::: "memory");
  __syncthreads();

  // ---- A fragments: chunks of H16 (entries +-1), built from lane id ----
  // A-matrix 16x4 f32 layout (ISA 7.12.2): lanes 0-15 -> K = 4c+0 (v0), 4c+1 (v1)
  //                                        lanes 16-31 -> K = 4c+2 (v0), 4c+3 (v1)
  const int m    = lane & 15;
  const int koff = (lane < 16) ? 0 : 2;
  v2f a[4];
#pragma unroll
  for (int c = 0; c < 4; ++c) {
    const int k0 = 4 * c + koff;
    a[c].x = (__builtin_popcount(m & k0)       & 1) ? -1.0f : 1.0f;
    a[c].y = (__builtin_popcount(m & (k0 + 1)) & 1) ? -1.0f : 1.0f;
  }

  const int n     = lane & 15;           // B column / D column per lane
  const int mhalf = (lane < 16) ? 0 : 8; // D row offset per lane half

  // Three passes: transform along k, j, i of the 16x16x16 view.
  // Tile element (r, c) of tile t sits at  t*st + r*sr + c*sc  (r = contraction axis).
  const int strides[3][3] = { {256, 1, 16},    // axis k: tile=i, cols=j
                              {256, 16, 1},    // axis j: tile=i, cols=k
                              {16, 256, 1} };  // axis i: tile=j, cols=k
#pragma unroll
  for (int p = 0; p < 3; ++p) {
    const int st = strides[p][0], sr = strides[p][1], sc = strides[p][2];
    for (int t = 0; t < 16; ++t) {
      const int tb = t * st;
      v8f acc = {};
#pragma unroll
      for (int c = 0; c < 4; ++c) {
        // B-matrix 4x16 f32: VGPR v, lane l -> K = 4c + koff + v, N = l&15
        const int kb = 4 * c + koff;
        v2f b;
        b.x = L[PADIDX(tb + kb * sr + n * sc)];
        b.y = L[PADIDX(tb + (kb + 1) * sr + n * sc)];
        acc = __builtin_amdgcn_wmma_f32_16x16x4_f32(
            /*neg_a=*/false, a[c], /*neg_b=*/false, b,
            /*c_mod=*/(short)0, acc, /*reuse_a=*/false, /*reuse_b=*/false);
      }
      // D 16x16 f32 store (ISA 7.12.2): VGPR g -> M = g + (lane<16 ? 0 : 8), N = lane&15
#pragma unroll
      for (int g = 0; g < 8; ++g) {
        L[PADIDX(tb + (g + mhalf) * sr + n * sc)] = acc[g];
      }
    }
    __syncthreads();
  }

  // ---- write transformed row back: async LDS -> global DMA ----
  if (valid) {
#pragma unroll
    for (int i = 0; i < FWHT_D / (32 * 4); ++i) {
      const int e = (i * 32 + lane) * 4;
      const uint32_t srcl = (uint32_t)(uintptr_t)(L + PADIDX(e));
      const uint64_t dst  = (uint64_t)(uintptr_t)(yr + e);
      asm volatile("global_store_async_from_lds_b128 %0, %1, off"
                   :: "v"(dst), "v"(srcl) : "memory");
    }
  }
  asm volatile("s_wait_asynccnt 0x0" ::: "memory");
}

extern "C" void kernel_launch(void* const* d_in, const int* in_sizes, int n_in,
                              void* d_out, int out_size, void* d_ws, size_t ws_size,
                              hipStream_t stream) {
  const float* x = (const float*)d_in[0];
  float*       y = (float*)d_out;
  const int nrows = in_sizes[0] / FWHT_D;        // 16384
  const int nblocks = (nrows + FWHT_WAVES - 1) / FWHT_WAVES;
  fwht4096_wmma_kernel<<<nblocks, FWHT_TPB, 0, stream>>>(x, y, nrows);
}